// CustomEmbedding_43224550867805
// MI455X (gfx1250) — compile-verified
//
#include <hip/hip_runtime.h>
#include <hip/hip_bf16.h>

// Embedding gather: out[i, :] = weight[:, idx[i]]  (weight is [D=128, V=100000])
// 1) TDM (tensor_load_to_lds) DMAs a strided 64x128 tile of weight into LDS with
//    hardware row padding (64+1 DWORDs -> bank-conflict-free), then waves write
//    the transposed tile to table[V][D] in workspace (table stays L2-resident).
// 2) One wave32 per row: lane l b128-loads table[row][4l..4l+3] (512B coalesced,
//    L2 hit) and streams it out with a non-temporal b128 store.

#define EMB_V 100000
#define EMB_D 128

typedef float    v4f   __attribute__((ext_vector_type(4)));
typedef unsigned u32x4 __attribute__((ext_vector_type(4)));
typedef int      i32x8 __attribute__((ext_vector_type(8)));
typedef int      i32x4 __attribute__((ext_vector_type(4)));

// ---------------------------------------------------------------------------
// Kernel 1: transpose weight[D][V] -> table[V][D], tile = 128 d x 64 v.
// Load phase is a single Tensor Data Mover descriptor per workgroup:
//   2D tile, tile_dim0=64 (x along V), tile_dim1=128 (rows along D),
//   tensor_dim0_stride=V, pad: +1 DWORD after every 64 DWORDs -> LDS row
//   stride 65 (conflict-free on 64 banks for the d-contiguous read-out).
//   tensor_dim0 = V - v0 gives OOB zero-fill on the ragged last tile.
// ---------------------------------------------------------------------------
__global__ __launch_bounds__(256) void emb_transpose_tdm_kernel(
    const float* __restrict__ w, float* __restrict__ table) {
  // Only LDS object in this kernel -> group-segment offset 0 (used as D#.lds_addr).
  __shared__ float tile[EMB_D * 65];  // 33,280 B: 128 rows of 64 data + 1 pad DWORD

  const int v0  = blockIdx.x * 64;
  const int tid = threadIdx.x;

  if (tid < 32) {  // wave 0 only (TDM ignores EXEC, so gate at wave granularity)
    const unsigned long long gaddr = (unsigned long long)(const void*)(w + v0);
    const unsigned td0 = (unsigned)(EMB_V - v0);  // x-extent bound for OOB zero-fill

    // ---- D# group 0 (4 SGPRs) ----
    u32x4 g0;
    g0.x = 1u;                                         // count=1, user desc, no gather
    g0.y = 0u;                                         // lds_addr = 0 (tile offset)
    g0.z = (unsigned)(gaddr & 0xFFFFFFFFull);          // global_addr[31:0]
    g0.w = (unsigned)((gaddr >> 32) & 0x01FFFFFFull)   // global_addr[56:32]
         | (2u << 30);                                 // type = 2 ("image")

    // ---- D# group 1 (8 SGPRs) ----
    i32x8 g1;
    g1[0] = (int)((2u << 16)        // data_size = 4 bytes
                | (1u << 20)        // pad_enable
                | (5u << 22)        // pad_interval: 64 DWORDs
                | (0u << 25));      // pad_amount: 1 DWORD
    g1[1] = (int)((td0 & 0xFFFFu) << 16);              // barrier_addr=0 | tensor_dim0[15:0]
    g1[2] = (int)((td0 >> 16) | (128u << 16));         // tensor_dim0[31:16] | tensor_dim1[15:0]=128
    g1[3] = (int)(64u << 16);                          // tensor_dim1[31:16]=0 | tile_dim0=64
    g1[4] = (int)128u;                                 // tile_dim1=128, tile_dim2=0
    g1[5] = (int)100000u;                              // tensor_dim0_stride[31:0] = V
    g1[6] = 0;                                         // stride[47:32]=0 | dim1_stride[15:0]=0
    g1[7] = 0;                                         // dim1_stride[47:16]=0

    const i32x4 gz4 = {0, 0, 0, 0};                    // groups 2/3: unused (2D tensor)
    const i32x8 gz8 = {0, 0, 0, 0, 0, 0, 0, 0};        // extra group (6-arg toolchain)

    __builtin_amdgcn_tensor_load_to_lds(g0, g1, gz4, gz4, gz8, /*cpol=*/0);
    __builtin_amdgcn_s_wait_tensorcnt((short)0);       // s_wait_tensorcnt 0
  }
  __syncthreads();  // waves 1..7 wait until wave 0 passed the tensorcnt wait

  // The TDM wrote `tile` behind the compiler's back (LDS address travels as an
  // integer inside the descriptor). Escape the pointer + full memory clobber so
  // the LDS reads below aren't folded to undef and the store loop isn't DCE'd.
  asm volatile("" : : "r"(&tile[0]) : "memory");

  // Store phase: 256 threads = 2 v x 128 consecutive d per step.
  // Global writes coalesced (512B per v-row); LDS reads stride-65 -> conflict-free.
  const int sd = tid & 127;  // d
  const int sv = tid >> 7;   // starting v (0..1)
  #pragma unroll 4
  for (int vv = sv; vv < 64; vv += 2) {
    if (v0 + vv < EMB_V) {
      table[(size_t)(v0 + vv) * EMB_D + sd] = tile[sd * 65 + vv];
    }
  }
}

// ---------------------------------------------------------------------------
// Kernel 2: gather. One wave32 per row per iteration:
//   lane l: b128 load of table[row] + 16*l (L2 hit), NT b128 store to out.
// Index is wave-uniform: readfirstlane -> SMEM-friendly scalar fetch.
// ---------------------------------------------------------------------------
__global__ __launch_bounds__(256) void emb_gather_kernel(
    const int* __restrict__ idx, const float* __restrict__ table,
    float* __restrict__ out, int n) {
  const int lane  = threadIdx.x & 31;
  const int wave  = (int)((blockIdx.x * blockDim.x + threadIdx.x) >> 5);
  const int nwave = (int)((gridDim.x * blockDim.x) >> 5);

  for (int i = wave; i < n; i += nwave) {
    // Run the index stream ahead of the loop (speculative prefetch; OOB is safe).
    __builtin_prefetch(&idx[i + nwave], 0, 0);

    const int i_s = __builtin_amdgcn_readfirstlane(i);  // provably uniform address
    const int row = idx[i_s];

    const v4f* __restrict__ src =
        reinterpret_cast<const v4f*>(table + (size_t)row * EMB_D);
    v4f val = src[lane];  // 32 lanes x 16B = 512B coalesced from L2-resident table

    v4f* __restrict__ dst = reinterpret_cast<v4f*>(out + (size_t)i * EMB_D);
    __builtin_nontemporal_store(val, dst + lane);  // 419MB stream: keep out of L2
  }
}

// ---------------------------------------------------------------------------
// Fallback (only if workspace can't hold the 51.2MB table): direct strided
// gather from weight (relies on weight fitting in L2).
// ---------------------------------------------------------------------------
__global__ __launch_bounds__(256) void emb_gather_direct_kernel(
    const int* __restrict__ idx, const float* __restrict__ w,
    float* __restrict__ out, long long total) {
  long long g = (long long)blockIdx.x * blockDim.x + threadIdx.x;
  const long long stride = (long long)gridDim.x * blockDim.x;
  for (; g < total; g += stride) {
    const long long i = g >> 7;   // row index position
    const int d = (int)(g & 127); // dim
    const int row = idx[i];
    __builtin_nontemporal_store(w[(size_t)d * EMB_V + row], &out[g]);
  }
}

extern "C" void kernel_launch(void* const* d_in, const int* in_sizes, int n_in,
                              void* d_out, int out_size, void* d_ws, size_t ws_size,
                              hipStream_t stream) {
  const int*   idx = (const int*)d_in[0];     // index: 4096*200 = 819200 entries
  const float* w   = (const float*)d_in[1];   // weight: [128, 100000] f32
  float*       out = (float*)d_out;           // [819200, 128] f32

  const int n = in_sizes[0];                  // 819200
  const size_t table_bytes = (size_t)EMB_V * EMB_D * sizeof(float);  // 51.2 MB

  if (ws_size >= table_bytes) {
    float* table = (float*)d_ws;
    emb_transpose_tdm_kernel<<<(EMB_V + 63) / 64, 256, 0, stream>>>(w, table);
    // 2048 blocks * 8 waves = 16384 waves; ~50 rows each — enough parallelism to
    // saturate HBM write bandwidth while reads hit L2.
    emb_gather_kernel<<<2048, 256, 0, stream>>>(idx, table, out, n);
  } else {
    emb_gather_direct_kernel<<<4096, 256, 0, stream>>>(
        idx, w, out, (long long)n * EMB_D);
  }
}